// TopKDecoder_52982716564242
// MI455X (gfx1250) — compile-verified
//
#include <hip/hip_runtime.h>
#include <hip/hip_bf16.h>
#include <math.h>

#define KBEAM 4
#define VOCAB 32000
#define HDIM  512
#define BATCH 16
#define LSEQ  64
#define TSTEPS 32
#define BK (BATCH*KBEAM)          // 64
#define SOS_ID 1
#define EOS_ID 2
#define INF_VAL 100000.0f
#define KC 128                    // K-chunk staged in LDS (KC x 16 floats = 8KB)

typedef __attribute__((ext_vector_type(2))) float v2f;
typedef __attribute__((ext_vector_type(4))) float v4f;
typedef __attribute__((ext_vector_type(8))) float v8f;
typedef __attribute__((ext_vector_type(4))) unsigned int v4u;
typedef __attribute__((ext_vector_type(4))) int v4i;
typedef __attribute__((ext_vector_type(8))) int v8i;

#if defined(__has_builtin)
#if __has_builtin(__builtin_amdgcn_tensor_load_to_lds)
#define HAVE_TDM 1
#endif
#endif

#ifdef HAVE_TDM
// Issue one TDM 2D tile load: rows x 16 f32 tile of B (row stride ldb elements)
// -> contiguous LDS at byte offset lds_off. D# layout per CDNA5 ISA §8.
__device__ __forceinline__ void tdm_load_panel(const float* gsrc, int ldb,
                                               unsigned lds_off, int rows) {
  unsigned long long ga = (unsigned long long)(uintptr_t)gsrc;
  v4u g0 = { 1u,                                    // count=1, user mode
             lds_off,                               // lds_addr (bytes)
             (unsigned)(ga & 0xffffffffu),          // global_addr[31:0]
             (unsigned)((ga >> 32) & 0x1ffffffu) | (2u << 30) }; // addr[56:32] | type=2
  v8i g1 = { (int)0x00020000,                       // data_size=2 (4B), mask=0
             (int)(16u << 16),                      // tensor_dim0 = 16 (bits 79:48 lo)
             (int)((unsigned)rows << 16),           // dim0 hi=0 | tensor_dim1 = rows
             (int)(16u << 16),                      // dim1 hi=0 | tile_dim0 = 16
             rows,                                  // tile_dim1 = rows | tile_dim2=0
             ldb,                                   // tensor_dim0_stride[31:0]
             0, 0 };                                // stride hi / dim1_stride = 0
  v4i gz = {0, 0, 0, 0};
#if defined(__clang_major__) && (__clang_major__ >= 23)
  v8i gz8 = {0, 0, 0, 0, 0, 0, 0, 0};
  __builtin_amdgcn_tensor_load_to_lds(g0, g1, gz, gz, gz8, 0);
#else
  __builtin_amdgcn_tensor_load_to_lds(g0, g1, gz, gz, 0);
#endif
}
#endif

// ---------------------------------------------------------------------------
// WMMA FP32 GEMM: C[64,N] = act(A[64,K] @ B[K,N] + bias)
// Block: 4 waves (wave w -> M rows [16w,16w+16)), blockIdx.x -> 16-col N tile.
// B panel staged in LDS (TDM double-buffered), shared by all 4 waves: cuts the
// dominant Wo stream 4x vs per-wave global loads. A (<=256KB) stays in cache.
// ---------------------------------------------------------------------------
__global__ __launch_bounds__(128) void wmma_gemm64_kernel(
    const float* __restrict__ A, int lda,
    const float* __restrict__ B, int ldb,
    float* __restrict__ C, int ldc,
    int N, int Kdim, const float* __restrict__ bias, int act) {
  __shared__ float bp[2][KC * 16];
  const int tid    = threadIdx.x;
  const int lane   = tid & 31;
  const int wave   = tid >> 5;           // 0..3 -> M tile
  const int half   = lane >> 4;          // 0 | 1 (K pair select)
  const int lane16 = lane & 15;
  const int m0 = wave * 16;
  const int n0 = blockIdx.x * 16;
  const int nch = Kdim / KC;
  if (n0 >= N) return;                   // wave-uniform

  const float* Arow = A + (size_t)(m0 + lane16) * lda;   // lane = M row
  const float* Bblk = B + n0;

#ifdef HAVE_TDM
  if (wave == 0)
    tdm_load_panel(Bblk, ldb, (unsigned)(uintptr_t)&bp[0][0], KC);
#endif

  v8f acc = {};
  for (int c = 0; c < nch; ++c) {
#ifdef HAVE_TDM
    if (wave == 0) {
      if (c + 1 < nch) {
        tdm_load_panel(Bblk + (size_t)(c + 1) * KC * ldb, ldb,
                       (unsigned)(uintptr_t)&bp[(c + 1) & 1][0], KC);
        __builtin_amdgcn_s_wait_tensorcnt(1);   // chunk c complete (in-order)
      } else {
        __builtin_amdgcn_s_wait_tensorcnt(0);
      }
    }
    __syncthreads();
#else
    {  // fallback: cooperative vectorized copy of chunk c
      const float* src = Bblk + (size_t)c * KC * ldb;
      float* dst = &bp[c & 1][0];
      for (int i = tid; i < KC * 16 / 4; i += 128) {
        int r = (i * 4) >> 4, col = (i * 4) & 15;
        *(v4f*)(dst + r * 16 + col) = *(const v4f*)(src + (size_t)r * ldb + col);
      }
      __syncthreads();
    }
#endif
    const float* bpc = &bp[c & 1][0];
    const float* Ac  = Arow + c * KC;
    #pragma unroll 8
    for (int kk = 0; kk < KC; kk += 4) {
      const int ka = kk + half * 2;
      v2f a = *(const v2f*)(Ac + ka);              // A[m][ka], A[m][ka+1]
      v2f b;
      b.x = bpc[(ka + 0) * 16 + lane16];           // B[ka][n]   (LDS, bank-clean)
      b.y = bpc[(ka + 1) * 16 + lane16];
      acc = __builtin_amdgcn_wmma_f32_16x16x4_f32(false, a, false, b,
                                                  (short)0, acc, false, false);
    }
    __syncthreads();                               // protect buffer reuse
  }

  const float bb = bias ? bias[n0 + lane16] : 0.0f;
  union { v8f v; float f[8]; } u; u.v = acc;
  #pragma unroll
  for (int r = 0; r < 8; ++r) {
    const int row = m0 + r + half * 8;             // C layout: VGPR r -> M=r (+8 hi half)
    float val = u.f[r] + bb;
    if (act == 1) val = tanhf(val);
    C[(size_t)row * ldc + n0 + lane16] = val;
  }
}

// ---------------------------------------------------------------------------
__global__ void init_kernel(const float* __restrict__ ehid, float* __restrict__ h,
                            int* __restrict__ inp, float* __restrict__ seq,
                            int* __restrict__ eos, float* __restrict__ eossc) {
  int idx = blockIdx.x * 256 + threadIdx.x;
  if (idx < BK * HDIM) {
    int beam = idx / HDIM;
    h[idx] = ehid[(beam / KBEAM) * HDIM + (idx % HDIM)];  // repeat per-batch hidden
  }
  if (idx < BK) {
    inp[idx]   = SOS_ID;
    seq[idx]   = (idx % KBEAM == 0) ? 0.0f : -1000.0f;
    eos[idx]   = 0;
    eossc[idx] = 0.0f;
  }
}

__global__ void embed_kernel(const float* __restrict__ emb, const int* __restrict__ inp,
                             float* __restrict__ x) {
  int idx = blockIdx.x * 256 + threadIdx.x;       // BK*HDIM
  if (idx >= BK * HDIM) return;
  int i = idx >> 9, c = idx & (HDIM - 1);
  x[idx] = emb[(size_t)inp[i] * HDIM + c];
}

__global__ void gru_update_kernel(const float* __restrict__ gx, const float* __restrict__ gh,
                                  const float* __restrict__ h, float* __restrict__ hnew,
                                  float* __restrict__ hc) {
  int idx = blockIdx.x * 256 + threadIdx.x;       // BK*HDIM
  if (idx >= BK * HDIM) return;
  int i = idx >> 9, c = idx & (HDIM - 1);
  const float xr = gx[i*3*HDIM + c],          hr = gh[i*3*HDIM + c];
  const float xz = gx[i*3*HDIM + HDIM + c],   hz = gh[i*3*HDIM + HDIM + c];
  const float xn = gx[i*3*HDIM + 2*HDIM + c], hn = gh[i*3*HDIM + 2*HDIM + c];
  const float r = 1.0f / (1.0f + __expf(-(xr + hr)));
  const float z = 1.0f / (1.0f + __expf(-(xz + hz)));
  const float n = tanhf(xn + r * hn);
  const float out = (1.0f - z) * n + z * h[idx];
  hnew[idx] = out;
  hc[(size_t)i * (2*HDIM) + c] = out;             // concat lower half
}

// att = softmax(h_new . enc), ctx = att @ enc ; one block per beam
__global__ __launch_bounds__(256) void attention_kernel(
    const float* __restrict__ hnew, const float* __restrict__ enc,
    float* __restrict__ hc) {
  const int beam = blockIdx.x;
  const float* E  = enc + (size_t)(beam / KBEAM) * LSEQ * HDIM;
  const float* hv = hnew + (size_t)beam * HDIM;
  __shared__ float sc[LSEQ];
  __shared__ float smax, ssum;
  const int tid = threadIdx.x, wave = tid >> 5, lane = tid & 31;
  for (int l = wave; l < LSEQ; l += 8) {
    float s = 0.0f;
    for (int c = lane; c < HDIM; c += 32) s += hv[c] * E[(size_t)l * HDIM + c];
    for (int off = 16; off > 0; off >>= 1) s += __shfl_down(s, off, 32);
    if (lane == 0) sc[l] = s;
  }
  __syncthreads();
  if (tid == 0) {
    float mx = sc[0];
    for (int l = 1; l < LSEQ; ++l) mx = fmaxf(mx, sc[l]);
    float sum = 0.0f;
    for (int l = 0; l < LSEQ; ++l) sum += __expf(sc[l] - mx);
    smax = mx; ssum = sum;
  }
  __syncthreads();
  for (int l = tid; l < LSEQ; l += 256) sc[l] = __expf(sc[l] - smax) / ssum;
  __syncthreads();
  for (int c = tid; c < HDIM; c += 256) {
    float acc = 0.0f;
    for (int l = 0; l < LSEQ; ++l) acc += sc[l] * E[(size_t)l * HDIM + c];
    hc[(size_t)beam * (2*HDIM) + HDIM + c] = acc;   // concat upper half
  }
}

// per-row log-softmax stats over VOCAB
__global__ __launch_bounds__(256) void rowstats_kernel(
    const float* __restrict__ logits, float* __restrict__ rowmax,
    float* __restrict__ rowlse) {
  const int r = blockIdx.x, tid = threadIdx.x;
  __shared__ float red[256];
  float mx = -3.4e38f;
  for (int v = tid; v < VOCAB; v += 256) mx = fmaxf(mx, logits[(size_t)r * VOCAB + v]);
  red[tid] = mx; __syncthreads();
  for (int s = 128; s > 0; s >>= 1) {
    if (tid < s) red[tid] = fmaxf(red[tid], red[tid + s]);
    __syncthreads();
  }
  mx = red[0]; __syncthreads();
  float sum = 0.0f;
  for (int v = tid; v < VOCAB; v += 256) sum += __expf(logits[(size_t)r * VOCAB + v] - mx);
  red[tid] = sum; __syncthreads();
  for (int s = 128; s > 0; s >>= 1) {
    if (tid < s) red[tid] += red[tid + s];
    __syncthreads();
  }
  if (tid == 0) { rowmax[r] = mx; rowlse[r] = logf(red[0]); }
}

// top-4 of K*V candidates per batch; updates beam state + history
__global__ __launch_bounds__(256) void topk_kernel(
    const float* __restrict__ logits, const float* __restrict__ rowmax,
    const float* __restrict__ rowlse, float* seq, const float* __restrict__ mask,
    int* eos, float* eossc, int* __restrict__ inp, int* __restrict__ pred,
    int* __restrict__ preds_h, int* __restrict__ syms_h,
    float* __restrict__ scores_h, float* __restrict__ saved_h, int t) {
  const int b = blockIdx.x, tid = threadIdx.x;
  float bv[KBEAM]; int bidx[KBEAM];
  #pragma unroll
  for (int j = 0; j < KBEAM; ++j) { bv[j] = -3.4e38f; bidx[j] = 0x7fffffff; }
  for (int idx = tid; idx < KBEAM * VOCAB; idx += 256) {
    const int rloc = idx / VOCAB, v = idx - rloc * VOCAB;
    const int r = b * KBEAM + rloc;
    const float lp = logits[(size_t)r * VOCAB + v] - rowmax[r] - rowlse[r];
    float val = seq[r] + lp + mask[(size_t)r * VOCAB + v];
    if (v == EOS_ID && eos[r]) val = eossc[r];    // frozen EOS beams keep their score
    if (val > bv[KBEAM - 1]) {
      int p = KBEAM - 1;
      while (p > 0 && val > bv[p - 1]) { bv[p] = bv[p-1]; bidx[p] = bidx[p-1]; --p; }
      bv[p] = val; bidx[p] = idx;
    }
  }
  __shared__ float sv[256 * KBEAM];
  __shared__ int   si[256 * KBEAM];
  #pragma unroll
  for (int j = 0; j < KBEAM; ++j) { sv[tid*KBEAM + j] = bv[j]; si[tid*KBEAM + j] = bidx[j]; }
  __syncthreads();
  if (tid == 0) {
    for (int j = 0; j < KBEAM; ++j) {
      float best = -3.4e38f; int besti = 0x7fffffff; int bp2 = -1;
      for (int q = 0; q < 256 * KBEAM; ++q) {
        if (sv[q] > best || (sv[q] == best && si[q] < besti)) {
          best = sv[q]; besti = si[q]; bp2 = q;
        }
      }
      sv[bp2] = -3.4e38f;                          // remove winner
      const int i   = b * KBEAM + j;
      const int sym = besti % VOCAB;
      const int pr  = b * KBEAM + besti / VOCAB;   // pos_index + top_i//V
      inp[i] = sym; pred[i] = pr;
      preds_h[t*BK + i]  = pr;
      syms_h[t*BK + i]   = sym;
      scores_h[t*BK + i] = best;                   // stored before masked_fill
      saved_h[t*BK + i]  = logits[(size_t)pr * VOCAB + sym] - rowmax[pr] - rowlse[pr];
      const int e = (sym == EOS_ID);
      eos[i]   = e;
      eossc[i] = e ? best : 0.0f;
      seq[i]   = e ? -1000.0f : best;
    }
  }
}

__global__ void mask_update_kernel(const float* __restrict__ msrc, float* __restrict__ mdst,
                                   const int* __restrict__ pred, const int* __restrict__ inp) {
  long idx = (long)blockIdx.x * 256 + threadIdx.x;   // BK*VOCAB
  if (idx >= (long)BK * VOCAB) return;
  const int i = (int)(idx / VOCAB), v = (int)(idx - (long)i * VOCAB);
  float m = msrc[(size_t)pred[i] * VOCAB + v];
  if (v == inp[i]) m -= INF_VAL;
  if (v == EOS_ID) m = 0.0f;
  mdst[idx] = m;
}

__global__ void h_gather_kernel(const float* __restrict__ hnew, float* __restrict__ h,
                                const int* __restrict__ pred) {
  int idx = blockIdx.x * 256 + threadIdx.x;
  if (idx >= BK * HDIM) return;
  h[idx] = hnew[(size_t)pred[idx >> 9] * HDIM + (idx & (HDIM - 1))];
}

// reverse scan; each thread owns one beam pointer (no cross-lane deps)
__global__ void backtrace_kernel(const int* __restrict__ preds, const int* __restrict__ syms,
                                 const float* __restrict__ savedlp,
                                 const float* __restrict__ scores_h, float* __restrict__ out) {
  const int j = threadIdx.x;                      // 0..63
  int now = j;
  for (int t = TSTEPS - 1; t >= 0; --t) {
    out[t*BK + j]              = (float)syms[t*BK + now];   // out_syms
    out[TSTEPS*BK + t*BK + j]  = savedlp[t*BK + now];       // out_scores
    now = preds[t*BK + now];
  }
  out[2*TSTEPS*BK + j] = scores_h[(TSTEPS - 1)*BK + j];     // final_scores
}

// ---------------------------------------------------------------------------
extern "C" void kernel_launch(void* const* d_in, const int* in_sizes, int n_in,
                              void* d_out, int out_size, void* d_ws, size_t ws_size,
                              hipStream_t stream) {
  (void)in_sizes; (void)n_in; (void)out_size; (void)ws_size;
  const float* enc_hidden = (const float*)d_in[0];
  const float* enc_out    = (const float*)d_in[1];
  const float* embedding  = (const float*)d_in[2];
  const float* Wx         = (const float*)d_in[3];
  const float* Wh         = (const float*)d_in[4];
  const float* bg         = (const float*)d_in[5];
  const float* Wc         = (const float*)d_in[6];
  const float* Wo         = (const float*)d_in[7];
  float* out = (float*)d_out;

  char* p = (char*)d_ws;
  auto alloc = [&](size_t bytes) { void* q = (void*)p; p += (bytes + 255) & ~(size_t)255; return q; };
  float* h      = (float*)alloc((size_t)BK*HDIM*4);
  float* x      = (float*)alloc((size_t)BK*HDIM*4);
  float* hnew   = (float*)alloc((size_t)BK*HDIM*4);
  float* hc     = (float*)alloc((size_t)BK*2*HDIM*4);
  float* gx     = (float*)alloc((size_t)BK*3*HDIM*4);
  float* gh     = (float*)alloc((size_t)BK*3*HDIM*4);
  float* comb   = (float*)alloc((size_t)BK*HDIM*4);
  float* logits = (float*)alloc((size_t)BK*VOCAB*4);
  float* maskA  = (float*)alloc((size_t)BK*VOCAB*4);
  float* maskB  = (float*)alloc((size_t)BK*VOCAB*4);
  float* rowmax = (float*)alloc(BK*4);
  float* rowlse = (float*)alloc(BK*4);
  float* seq    = (float*)alloc(BK*4);
  float* eossc  = (float*)alloc(BK*4);
  int*   inp    = (int*)alloc(BK*4);
  int*   pred   = (int*)alloc(BK*4);
  int*   eos    = (int*)alloc(BK*4);
  int*   preds_h  = (int*)alloc((size_t)TSTEPS*BK*4);
  int*   syms_h   = (int*)alloc((size_t)TSTEPS*BK*4);
  float* scores_h = (float*)alloc((size_t)TSTEPS*BK*4);
  float* saved_h  = (float*)alloc((size_t)TSTEPS*BK*4);

  hipMemsetAsync(maskA, 0, (size_t)BK*VOCAB*4, stream);
  init_kernel<<<(BK*HDIM + 255)/256, 256, 0, stream>>>(enc_hidden, h, inp, seq, eos, eossc);

  float* mcur = maskA; float* mnxt = maskB;
  for (int t = 0; t < TSTEPS; ++t) {
    embed_kernel<<<(BK*HDIM + 255)/256, 256, 0, stream>>>(embedding, inp, x);
    // gx = x @ Wx + b ; gh = h @ Wh
    wmma_gemm64_kernel<<<(3*HDIM)/16, 128, 0, stream>>>(x, HDIM, Wx, 3*HDIM, gx, 3*HDIM,
                                                        3*HDIM, HDIM, bg, 0);
    wmma_gemm64_kernel<<<(3*HDIM)/16, 128, 0, stream>>>(h, HDIM, Wh, 3*HDIM, gh, 3*HDIM,
                                                        3*HDIM, HDIM, nullptr, 0);
    gru_update_kernel<<<(BK*HDIM + 255)/256, 256, 0, stream>>>(gx, gh, h, hnew, hc);
    attention_kernel<<<BK, 256, 0, stream>>>(hnew, enc_out, hc);
    // comb = tanh(hc @ Wc)
    wmma_gemm64_kernel<<<HDIM/16, 128, 0, stream>>>(hc, 2*HDIM, Wc, HDIM, comb, HDIM,
                                                    HDIM, 2*HDIM, nullptr, 1);
    // logits = comb @ Wo   (Wo is L2-resident after first step: 65.5MB < 192MB)
    wmma_gemm64_kernel<<<VOCAB/16, 128, 0, stream>>>(comb, HDIM, Wo, VOCAB, logits, VOCAB,
                                                     VOCAB, HDIM, nullptr, 0);
    rowstats_kernel<<<BK, 256, 0, stream>>>(logits, rowmax, rowlse);
    topk_kernel<<<BATCH, 256, 0, stream>>>(logits, rowmax, rowlse, seq, mcur, eos, eossc,
                                           inp, pred, preds_h, syms_h, scores_h, saved_h, t);
    mask_update_kernel<<<(int)(((size_t)BK*VOCAB + 255)/256), 256, 0, stream>>>(mcur, mnxt, pred, inp);
    h_gather_kernel<<<(BK*HDIM + 255)/256, 256, 0, stream>>>(hnew, h, pred);
    float* tmp = mcur; mcur = mnxt; mnxt = tmp;
  }
  backtrace_kernel<<<1, BK, 0, stream>>>(preds_h, syms_h, saved_h, scores_h, out);
}